// DefogFilter_16518444220681
// MI455X (gfx1250) — compile-verified
//
#include <hip/hip_runtime.h>
#include <math.h>

// ---------------------------------------------------------------------------
// DefogFilter (dark-channel haze removal) for MI455X / gfx1250.
// B=32 images, 3x512x512 f32. Memory-bound: ~200MB roofline traffic (~9us).
// Exact top-262 selection via 2-level radix histogram on float bit patterns.
// d_out is reused as 48MB scratch (dark bits + 2 histograms) before pass 7
// fully overwrites it. d_ws holds only ~132KB of per-image metadata.
// CDNA5 paths: global_load_async_to_lds_b128 (ASYNCcnt) staging in pass 1,
// s_wait_asynccnt, NT-hinted b128 streaming + global_prefetch_b8 in pass 7.
// ---------------------------------------------------------------------------

#define IMG_B   32
#define IMG_HW  (512 * 512)          // 262144 pixels per image
#define NUMPX   262                  // (512*512)//1000
#define NBIN    65536
#define EQCAP   1024
#define META_STRIDE 1056             // u32 per image: 32 header + 1024 eq-list

// u32 offsets inside d_out used as scratch (total 12,582,912 u32 < 25,165,824)
#define H1_OFF  ((size_t)IMG_B * (size_t)IMG_HW)
#define H2_OFF  (H1_OFF + (size_t)IMG_B * (size_t)NBIN)

// meta header slots (u32 / f32 union view)
//  [0]=B1  [1]=cntAbove1  [2]=V cutoff bits  [3]=needEq  [4]=eqCount
//  [8..10]=sumRGB(f32 atomics) [11..13]=A [14..16]=1/A [17]=w  [32..)=eqList

#define AS1 __attribute__((address_space(1)))
#define AS3 __attribute__((address_space(3)))

typedef float    v4f __attribute__((ext_vector_type(4)));
typedef unsigned v4u __attribute__((ext_vector_type(4)));
typedef int      v4i __attribute__((ext_vector_type(4)));

#if defined(__HIP_DEVICE_COMPILE__) && defined(__gfx1250__) && \
    __has_builtin(__builtin_amdgcn_global_load_async_to_lds_b128)
#define USE_ASYNC_LDS 1
#else
#define USE_ASYNC_LDS 0
#endif

__device__ __forceinline__ void wait_async_zero() {
#if defined(__HIP_DEVICE_COMPILE__) && defined(__gfx1250__)
#if __has_builtin(__builtin_amdgcn_s_wait_asynccnt)
  __builtin_amdgcn_s_wait_asynccnt(0);
#else
  asm volatile("s_wait_asynccnt 0" ::: "memory");
#endif
#endif
}

__global__ void k_zero_u32(unsigned* __restrict__ p, int n) {
  int i = blockIdx.x * blockDim.x + threadIdx.x;
  if (i < n) p[i] = 0u;
}

// ---- Pass 1: dark channel (stored as monotonic u32 bits) + coarse hist ----
__global__ void k_dark_hist(const float* __restrict__ img,
                            unsigned* __restrict__ scr) {
  const int bpi = IMG_HW / 1024;                 // 256 blocks / image
  const int i   = blockIdx.x / bpi;
  const int t   = threadIdx.x;                   // 0..255
  const int p4  = (blockIdx.x % bpi) * 256 + t;  // float4 index in image
  const float* basep = img + (size_t)i * 3 * IMG_HW;

  v4f r4, g4, b4;
#if USE_ASYNC_LDS
  __shared__ alignas(16) float shR[1024];
  __shared__ alignas(16) float shG[1024];
  __shared__ alignas(16) float shB[1024];
  // Stage the three color-plane tiles into LDS with the async copy engine.
  __builtin_amdgcn_global_load_async_to_lds_b128(
      (AS1 v4i*)(basep + (size_t)p4 * 4),
      (AS3 v4i*)&shR[t * 4], 0, 0);
  __builtin_amdgcn_global_load_async_to_lds_b128(
      (AS1 v4i*)(basep + IMG_HW + (size_t)p4 * 4),
      (AS3 v4i*)&shG[t * 4], 0, 0);
  __builtin_amdgcn_global_load_async_to_lds_b128(
      (AS1 v4i*)(basep + 2 * IMG_HW + (size_t)p4 * 4),
      (AS3 v4i*)&shB[t * 4], 0, 0);
  wait_async_zero();                             // ASYNCcnt == 0
  r4 = ((const v4f*)shR)[t];
  g4 = ((const v4f*)shG)[t];
  b4 = ((const v4f*)shB)[t];
#else
  r4 = ((const v4f*)basep)[p4];
  g4 = ((const v4f*)(basep + IMG_HW))[p4];
  b4 = ((const v4f*)(basep + 2 * IMG_HW))[p4];
#endif

  v4u d;
  d.x = __float_as_uint(fminf(fminf(r4.x, g4.x), b4.x));
  d.y = __float_as_uint(fminf(fminf(r4.y, g4.y), b4.y));
  d.z = __float_as_uint(fminf(fminf(r4.z, g4.z), b4.z));
  d.w = __float_as_uint(fminf(fminf(r4.w, g4.w), b4.w));

  ((v4u*)(scr + (size_t)i * IMG_HW))[p4] = d;    // cache dark bits in d_out

  unsigned* h1 = scr + H1_OFF + (size_t)i * NBIN;
  atomicAdd(&h1[d.x >> 16], 1u);
  atomicAdd(&h1[d.y >> 16], 1u);
  atomicAdd(&h1[d.z >> 16], 1u);
  atomicAdd(&h1[d.w >> 16], 1u);
}

// ---- Pass 2/4: per-image suffix-scan of a 65536-bin histogram -------------
// Find the largest bin b with cum-from-top(b) >= T.
__global__ void k_scan(const unsigned* __restrict__ hist,
                       unsigned* __restrict__ meta, int phase) {
  const int i = blockIdx.x;
  unsigned* mu = meta + (size_t)i * META_STRIDE;
  const unsigned T = (phase == 1) ? (unsigned)NUMPX
                                  : (unsigned)NUMPX - mu[1];
  const unsigned* h  = hist + (size_t)i * NBIN;
  const int t = threadIdx.x;                     // 256 threads, 256 bins each
  const unsigned* hb = h + t * 256;

  __shared__ unsigned suf[256];
  unsigned s = 0;
  for (int j = 0; j < 256; ++j) s += hb[j];
  suf[t] = s;
  __syncthreads();
  for (int off = 1; off < 256; off <<= 1) {      // Kogge-Stone suffix sum
    unsigned add = (t + off < 256) ? suf[t + off] : 0u;
    __syncthreads();
    suf[t] += add;
    __syncthreads();
  }
  const unsigned mysuf = suf[t];
  const unsigned above = (t < 255) ? suf[t + 1] : 0u;
  if (mysuf >= T && above < T) {                 // exactly one thread
    unsigned c = above;
    for (int j = 255; j >= 0; --j) {
      const unsigned hv = hb[j];
      if (c + hv >= T) {
        const unsigned bin = (unsigned)(t * 256 + j);
        if (phase == 1) { mu[0] = bin; mu[1] = c; }
        else           { mu[2] = (mu[0] << 16) | bin; mu[3] = T - c; }
        break;
      }
      c += hv;
    }
  }
}

// ---- Pass 3: fine histogram restricted to coarse bin B1 -------------------
__global__ void k_hist2(const unsigned* __restrict__ dark,
                        unsigned* __restrict__ hist2,
                        const unsigned* __restrict__ meta) {
  const int bpi = IMG_HW / 1024;
  const int i   = blockIdx.x / bpi;
  const int p4  = (blockIdx.x % bpi) * 256 + threadIdx.x;
  const unsigned B1 = meta[(size_t)i * META_STRIDE + 0];
  const v4u u = ((const v4u*)(dark + (size_t)i * IMG_HW))[p4];
  unsigned* h2 = hist2 + (size_t)i * NBIN;
  if ((u.x >> 16) == B1) atomicAdd(&h2[u.x & 0xFFFFu], 1u);
  if ((u.y >> 16) == B1) atomicAdd(&h2[u.y & 0xFFFFu], 1u);
  if ((u.z >> 16) == B1) atomicAdd(&h2[u.z & 0xFFFFu], 1u);
  if ((u.w >> 16) == B1) atomicAdd(&h2[u.w & 0xFFFFu], 1u);
}

// ---- Pass 5: sum RGB of pixels strictly above cutoff; collect equals ------
__global__ void k_select_sum(const unsigned* __restrict__ dark,
                             const float* __restrict__ img,
                             unsigned* __restrict__ meta) {
  const int bpi = IMG_HW / 1024;
  const int i   = blockIdx.x / bpi;
  const int p4  = (blockIdx.x % bpi) * 256 + threadIdx.x;
  unsigned* mu = meta + (size_t)i * META_STRIDE;
  const unsigned V = mu[2];
  const v4u u = ((const v4u*)(dark + (size_t)i * IMG_HW))[p4];
  const float* basep = img + (size_t)i * 3 * IMG_HW;
  float* mf = (float*)mu;
  const unsigned uv[4] = {u.x, u.y, u.z, u.w};
#pragma unroll
  for (int k = 0; k < 4; ++k) {
    const unsigned p = (unsigned)p4 * 4u + (unsigned)k;
    if (uv[k] > V) {
      atomicAdd(&mf[8],  basep[p]);
      atomicAdd(&mf[9],  basep[IMG_HW + p]);
      atomicAdd(&mf[10], basep[2 * IMG_HW + p]);
    } else if (uv[k] == V) {
      const unsigned pos = atomicAdd(&mu[4], 1u);
      if (pos < EQCAP) mu[32 + pos] = p;
    }
  }
}

// ---- Pass 6: tie-break equals by lowest index; finish A, 1/A, w -----------
__global__ void k_finalize(const float* __restrict__ img,
                           const float* __restrict__ param,
                           unsigned* __restrict__ meta) {
  const int i = blockIdx.x;
  if (threadIdx.x != 0) return;
  unsigned* mu = meta + (size_t)i * META_STRIDE;
  float* mf = (float*)mu;
  unsigned need = mu[3];
  unsigned m = mu[4]; if (m > EQCAP) m = EQCAP;
  if (need > m) need = m;
  unsigned* list = mu + 32;
  for (unsigned a = 0; a < need; ++a) {          // partial selection sort
    unsigned best = a;
    for (unsigned b = a + 1; b < m; ++b)
      if (list[b] < list[best]) best = b;
    unsigned tmp = list[a]; list[a] = list[best]; list[best] = tmp;
  }
  float sr = mf[8], sg = mf[9], sb = mf[10];
  const float* basep = img + (size_t)i * 3 * IMG_HW;
  for (unsigned a = 0; a < need; ++a) {
    const unsigned p = list[a];
    sr += basep[p];
    sg += basep[IMG_HW + p];
    sb += basep[2 * IMG_HW + p];
  }
  const float inv = 1.0f / (float)NUMPX;
  const float Ar = sr * inv, Ag = sg * inv, Ab = sb * inv;
  mf[11] = Ar; mf[12] = Ag; mf[13] = Ab;
  mf[14] = 1.0f / Ar; mf[15] = 1.0f / Ag; mf[16] = 1.0f / Ab;
  const float th = tanhf(param[i]) * 0.5f + 0.5f;
  mf[17] = th * 0.9f + 0.1f;                     // tanh_range(0.1, 1.0)
}

// ---- Pass 7: streaming recovery (fully overwrites d_out) ------------------
__global__ void k_recover(const float* __restrict__ img,
                          const unsigned* __restrict__ meta,
                          float* __restrict__ out) {
  const int bpi = IMG_HW / 1024;
  const int i   = blockIdx.x / bpi;
  const int p4  = (blockIdx.x % bpi) * 256 + threadIdx.x;
  const float* mf = (const float*)(meta + (size_t)i * META_STRIDE);
  const float Ar = mf[11], Ag = mf[12], Ab = mf[13];
  const float iAr = mf[14], iAg = mf[15], iAb = mf[16];
  const float w = mf[17];
  const float* basep = img + (size_t)i * 3 * IMG_HW;

  __builtin_prefetch(basep + (size_t)p4 * 4 + 2048, 0, 1);  // global_prefetch_b8

  const v4f r4 = __builtin_nontemporal_load((const v4f*)basep + p4);
  const v4f g4 = __builtin_nontemporal_load((const v4f*)(basep + IMG_HW) + p4);
  const v4f b4 = __builtin_nontemporal_load((const v4f*)(basep + 2 * IMG_HW) + p4);

  const float rr[4] = {r4.x, r4.y, r4.z, r4.w};
  const float gg[4] = {g4.x, g4.y, g4.z, g4.w};
  const float bb[4] = {b4.x, b4.y, b4.z, b4.w};
  float ro[4], go[4], bo[4];
#pragma unroll
  for (int k = 0; k < 4; ++k) {
    const float ica = fminf(fminf(rr[k] * iAr, gg[k] * iAg), bb[k] * iAb);
    const float t   = fmaxf(1.0f - w * ica, 0.01f);
    const float rt  = 1.0f / t;
    ro[k] = (rr[k] - Ar) * rt + Ar;
    go[k] = (gg[k] - Ag) * rt + Ag;
    bo[k] = (bb[k] - Ab) * rt + Ab;
  }
  const v4f orr = {ro[0], ro[1], ro[2], ro[3]};
  const v4f og  = {go[0], go[1], go[2], go[3]};
  const v4f ob  = {bo[0], bo[1], bo[2], bo[3]};

  float* obase = out + (size_t)i * 3 * IMG_HW;
  __builtin_nontemporal_store(orr, (v4f*)obase + p4);
  __builtin_nontemporal_store(og,  (v4f*)(obase + IMG_HW) + p4);
  __builtin_nontemporal_store(ob,  (v4f*)(obase + 2 * IMG_HW) + p4);
}

// ---------------------------------------------------------------------------
extern "C" void kernel_launch(void* const* d_in, const int* in_sizes, int n_in,
                              void* d_out, int out_size, void* d_ws,
                              size_t ws_size, hipStream_t stream) {
  (void)in_sizes; (void)n_in; (void)out_size; (void)ws_size;
  const float* img   = (const float*)d_in[0];
  const float* param = (const float*)d_in[1];
  float*    out  = (float*)d_out;
  unsigned* scr  = (unsigned*)d_out;             // scratch alias until pass 7
  unsigned* meta = (unsigned*)d_ws;              // ~132 KB of metadata

  const int nblk = IMG_B * (IMG_HW / 1024);      // 8192 blocks, 256 thr (8 waves)

  { int n = IMG_B * NBIN * 2;
    k_zero_u32<<<(n + 255) / 256, 256, 0, stream>>>(scr + H1_OFF, n); }
  { int n = IMG_B * META_STRIDE;
    k_zero_u32<<<(n + 255) / 256, 256, 0, stream>>>(meta, n); }

  k_dark_hist <<<nblk, 256, 0, stream>>>(img, scr);
  k_scan      <<<IMG_B, 256, 0, stream>>>(scr + H1_OFF, meta, 1);
  k_hist2     <<<nblk, 256, 0, stream>>>(scr, scr + H2_OFF, meta);
  k_scan      <<<IMG_B, 256, 0, stream>>>(scr + H2_OFF, meta, 2);
  k_select_sum<<<nblk, 256, 0, stream>>>(scr, img, meta);
  k_finalize  <<<IMG_B, 32, 0, stream>>>(img, param, meta);
  k_recover   <<<nblk, 256, 0, stream>>>(img, meta, out);
}